// ReductionByDistance_64656437674615
// MI455X (gfx1250) — compile-verified
//
#include <hip/hip_runtime.h>

// ReductionByDistance on MI455X (gfx1250, wave32).
// Pipeline: K0 norms -> K1 banded Gram via V_WMMA_F32_16X16X4_F32 + next[] ->
//           K2 pointer chase (segment starts) -> K3 contiguous-run means.
// K1 stages its 80x128 f32 panels with GLOBAL_LOAD_ASYNC_TO_LDS_B128
// (ASYNCcnt-tracked, no VGPR round-trip) when the builtin is available.

#define T_FRAMES 4096
#define DIMS     512
#define TILE     16
#define NTILES   5            // band of 5 tiles = 80 frames forward coverage
#define BAND     (NTILES * TILE)
#define KCHUNK   128          // K staged per LDS pass
#define LROW     (KCHUNK + 4) // LDS row stride (floats), bank-conflict padding
#define GROW     (BAND + 4)
#define CUTOFF2  1024.0f      // 32^2
#define WINDOW   64           // guaranteed forward search window per frame

typedef float v2f __attribute__((ext_vector_type(2)));
typedef float v8f __attribute__((ext_vector_type(8)));

#if __has_builtin(__builtin_amdgcn_wmma_f32_16x16x4_f32)
#define HAVE_WMMA_F32X4 1
#else
#define HAVE_WMMA_F32X4 0
#endif

#if __has_builtin(__builtin_amdgcn_global_load_async_to_lds_b128) && \
    __has_builtin(__builtin_amdgcn_s_wait_asynccnt)
#define HAVE_ASYNC_LDS 1
#else
#define HAVE_ASYNC_LDS 0
#endif

#if HAVE_ASYNC_LDS
typedef int v4i __attribute__((ext_vector_type(4)));
typedef __attribute__((address_space(1))) v4i g_v4i;   // global b128 pointer
typedef __attribute__((address_space(3))) v4i l_v4i;   // LDS b128 pointer
#endif

// ---------------- K0: squared norms of every frame ----------------
__global__ __launch_bounds__(32)
void k0_norms(const float* __restrict__ x, float* __restrict__ norms)
{
    const int t = blockIdx.x, s = blockIdx.y, lane = threadIdx.x;
    const float* row = x + ((size_t)s * T_FRAMES + t) * DIMS + lane * 16;
    float acc = 0.f;
#pragma unroll
    for (int i = 0; i < 4; ++i) {
        float4 v = ((const float4*)row)[i];
        acc += v.x * v.x + v.y * v.y + v.z * v.z + v.w * v.w;
    }
#pragma unroll
    for (int off = 16; off > 0; off >>= 1)
        acc += __shfl_down(acc, off, 32);
    if (lane == 0) norms[(size_t)s * T_FRAMES + t] = acc;
}

// ---------------- K1: banded Gram (WMMA) + next-boundary pointers ----------
__global__ __launch_bounds__(32)
void k1_next(const float* __restrict__ x, const float* __restrict__ norms,
             int* __restrict__ nxt)
{
    __shared__ float panel[BAND][LROW];   // 80 frames x 128 dims staged
    __shared__ float gband[TILE][GROW];   // G[m][c]: f_{j0+m} . f_{j0+c}
    __shared__ float nb[BAND];            // squared norms of band frames

    const int s    = blockIdx.y;
    const int j0   = blockIdx.x * TILE;
    const int lane = threadIdx.x;
    const float* base = x + (size_t)s * T_FRAMES * DIMS;

#if HAVE_WMMA_F32X4
    v8f acc[NTILES] = {};
#else
    for (int p = lane; p < TILE * GROW; p += 32) ((float*)gband)[p] = 0.f;
#endif

    for (int kb = 0; kb < DIMS / KCHUNK; ++kb) {
        // Stage BAND x KCHUNK panel: 32 lanes x 16B = one 512B row per issue.
#if HAVE_ASYNC_LDS
        for (int r = 0; r < BAND; ++r) {
            const int g = j0 + r;
            if (g < T_FRAMES) {
                // Direct memory->LDS, ASYNCcnt-tracked; no VGPR round trip.
                __builtin_amdgcn_global_load_async_to_lds_b128(
                    (g_v4i*)(base + (size_t)g * DIMS + kb * KCHUNK + lane * 4),
                    (l_v4i*)&panel[r][lane * 4], 0, 0);
            } else {
                *(float4*)&panel[r][lane * 4] = make_float4(0.f, 0.f, 0.f, 0.f);
            }
        }
        __builtin_amdgcn_s_wait_asynccnt(0);
#else
        for (int r = 0; r < BAND; ++r) {
            const int g = j0 + r;
            float4 v = make_float4(0.f, 0.f, 0.f, 0.f);
            if (g < T_FRAMES)
                v = *(const float4*)(base + (size_t)g * DIMS + kb * KCHUNK + lane * 4);
            *(float4*)&panel[r][lane * 4] = v;
        }
#endif
        __syncthreads();

#if HAVE_WMMA_F32X4
        // f32 A 16x4 layout: lanes 0-15 hold (K=k,k+1), lanes 16-31 (K=k+2,k+3),
        // row M = lane&15.  B 4x16 mirrors it with N = lane&15.
        const int row  = lane & 15;
        const int koff = (lane >> 4) << 1;
#pragma unroll 4
        for (int kc = 0; kc < KCHUNK; kc += 4) {
            v2f a = *(const v2f*)&panel[row][kc + koff];
#pragma unroll
            for (int t = 0; t < NTILES; ++t) {
                v2f b = *(const v2f*)&panel[t * TILE + row][kc + koff];
                acc[t] = __builtin_amdgcn_wmma_f32_16x16x4_f32(
                    false, a, false, b, (short)0, acc[t], false, false);
            }
        }
#else
        for (int p = lane; p < TILE * BAND; p += 32) {
            const int m = p / BAND, n = p % BAND;
            float d = 0.f;
            for (int k = 0; k < KCHUNK; ++k) d += panel[m][k] * panel[n][k];
            gband[m][n] += d;
        }
#endif
        __syncthreads();
    }

#if HAVE_WMMA_F32X4
    {   // Spill C-layout accumulators: VGPR v -> M = v + 8*(lane>=16), N = lane&15.
        const int Mbase = (lane >> 4) << 3;
        const int N     = lane & 15;
#pragma unroll
        for (int t = 0; t < NTILES; ++t)
#pragma unroll
            for (int v = 0; v < 8; ++v)
                gband[Mbase + v][t * TILE + N] = acc[t][v];
    }
#endif
    for (int i = lane; i < BAND; i += 32)
        nb[i] = (j0 + i < T_FRAMES) ? norms[(size_t)s * T_FRAMES + j0 + i] : 0.f;
    __syncthreads();

    // Each of the 16 low lanes scans one row of the band:
    // next(j) = first i in (j, j+WINDOW] with nj + ni - 2*G > cutoff^2.
    if (lane < TILE) {
        const int m = lane, j = j0 + m;
        const float nj = nb[m];
        int irelMax = m + WINDOW;                 // <= 79, always inside band
        const int lim = T_FRAMES - 1 - j0;
        if (irelMax > lim) irelMax = lim;
        int found = -2;
        for (int ir = m + 1; ir <= irelMax; ++ir) {
            const float d2 = nj + nb[ir] - 2.0f * gband[m][ir];
            if (d2 > CUTOFF2) { found = j0 + ir; break; }
        }
        if (found == -2)
            found = (j + WINDOW >= T_FRAMES - 1) ? T_FRAMES : -1; // -1: rare fallback
        nxt[(size_t)s * T_FRAMES + j] = found;
    }
}

// ---------------- K2: per-sequence chain chase -> segment starts -----------
__global__ __launch_bounds__(256)
void k2_chase(const float* __restrict__ x, const int* __restrict__ nxt,
              int* __restrict__ starts, int* __restrict__ nsegs,
              float* __restrict__ outTail)
{
    __shared__ int nl[T_FRAMES];
    const int s = blockIdx.x;
    for (int i = threadIdx.x; i < T_FRAMES; i += blockDim.x)
        nl[i] = nxt[(size_t)s * T_FRAMES + i];
    __syncthreads();
    if (threadIdx.x >= 32) return;    // wave 0 chases (LDS hops ~ segment count)

    const int   lane = threadIdx.x;
    const float* base = x + (size_t)s * T_FRAMES * DIMS;
    int* st = starts + (size_t)s * (T_FRAMES + 1);

    int cur = 0, k = 0;
    while (true) {
        if (lane == 0) st[k] = cur;
        ++k;
        int nx = nl[cur];                          // uniform LDS broadcast read
        if (nx == -1) {
            // Cooperative exact fallback (p ~ 2^-64): keep scanning past window.
            const float* rep = base + (size_t)cur * DIMS + lane * 16;
            float4 r0 = ((const float4*)rep)[0], r1 = ((const float4*)rep)[1];
            float4 r2 = ((const float4*)rep)[2], r3 = ((const float4*)rep)[3];
            nx = T_FRAMES;
            for (int i = cur + WINDOW + 1; i < T_FRAMES; ++i) {
                const float* f = base + (size_t)i * DIMS + lane * 16;
                float4 a0 = ((const float4*)f)[0], a1 = ((const float4*)f)[1];
                float4 a2 = ((const float4*)f)[2], a3 = ((const float4*)f)[3];
                float d = 0.f;
                d += (r0.x-a0.x)*(r0.x-a0.x) + (r0.y-a0.y)*(r0.y-a0.y)
                   + (r0.z-a0.z)*(r0.z-a0.z) + (r0.w-a0.w)*(r0.w-a0.w);
                d += (r1.x-a1.x)*(r1.x-a1.x) + (r1.y-a1.y)*(r1.y-a1.y)
                   + (r1.z-a1.z)*(r1.z-a1.z) + (r1.w-a1.w)*(r1.w-a1.w);
                d += (r2.x-a2.x)*(r2.x-a2.x) + (r2.y-a2.y)*(r2.y-a2.y)
                   + (r2.z-a2.z)*(r2.z-a2.z) + (r2.w-a2.w)*(r2.w-a2.w);
                d += (r3.x-a3.x)*(r3.x-a3.x) + (r3.y-a3.y)*(r3.y-a3.y)
                   + (r3.z-a3.z)*(r3.z-a3.z) + (r3.w-a3.w)*(r3.w-a3.w);
#pragma unroll
                for (int off = 16; off > 0; off >>= 1)
                    d += __shfl_xor(d, off, 32);
                if (d > CUTOFF2) { nx = i; break; }
            }
        }
        if (nx >= T_FRAMES) break;
        cur = nx;
        if (k > T_FRAMES) break;  // safety net
    }
    if (lane == 0) {
        st[k] = T_FRAMES;               // terminator for K3
        nsegs[s] = k;
        ((int*)outTail)[s] = k;         // n_segs output (int32 bit-pattern)
    }
}

// ---------------- K3: contiguous-run means + zero padding ------------------
__global__ __launch_bounds__(64)
void k3_means(const float* __restrict__ x, const int* __restrict__ starts,
              const int* __restrict__ nsegs, float* __restrict__ out)
{
    const int k = blockIdx.x, s = blockIdx.y, t = threadIdx.x;
    float* orow = out + ((size_t)s * T_FRAMES + k) * DIMS + t * 8;
    const int n = nsegs[s];
    if (k >= n) {   // harness poisons d_out: padded rows must be zeroed
        ((float4*)orow)[0] = make_float4(0.f, 0.f, 0.f, 0.f);
        ((float4*)orow)[1] = make_float4(0.f, 0.f, 0.f, 0.f);
        return;
    }
    const int* st = starts + (size_t)s * (T_FRAMES + 1);
    const int a = st[k], b = st[k + 1];
    const float* base = x + (size_t)s * T_FRAMES * DIMS + t * 8;
    float4 s0 = make_float4(0.f, 0.f, 0.f, 0.f);
    float4 s1 = make_float4(0.f, 0.f, 0.f, 0.f);
    for (int f = a; f < b; ++f) {
        if (f + 1 < b) __builtin_prefetch(base + (size_t)(f + 1) * DIMS, 0, 0);
        const float4* p = (const float4*)(base + (size_t)f * DIMS);
        float4 u = p[0], v = p[1];
        s0.x += u.x; s0.y += u.y; s0.z += u.z; s0.w += u.w;
        s1.x += v.x; s1.y += v.y; s1.z += v.z; s1.w += v.w;
    }
    const float inv = 1.0f / (float)(b - a);
    s0.x *= inv; s0.y *= inv; s0.z *= inv; s0.w *= inv;
    s1.x *= inv; s1.y *= inv; s1.z *= inv; s1.w *= inv;
    ((float4*)orow)[0] = s0;
    ((float4*)orow)[1] = s1;
}

extern "C" void kernel_launch(void* const* d_in, const int* in_sizes, int n_in,
                              void* d_out, int out_size, void* d_ws, size_t ws_size,
                              hipStream_t stream)
{
    (void)n_in; (void)out_size; (void)ws_size;
    const float* x = (const float*)d_in[0];
    float* out = (float*)d_out;
    const int B = in_sizes[0] / (T_FRAMES * DIMS);   // batch = 16

    // Workspace carve-out (~786 KB total).
    float* norms  = (float*)d_ws;                    // B*T floats
    int*   nxt    = (int*)(norms + (size_t)B * T_FRAMES);
    int*   starts = nxt + (size_t)B * T_FRAMES;      // B*(T+1) ints
    int*   nsegs  = starts + (size_t)B * (T_FRAMES + 1);

    k0_norms<<<dim3(T_FRAMES, B), 32, 0, stream>>>(x, norms);
    k1_next <<<dim3(T_FRAMES / TILE, B), 32, 0, stream>>>(x, norms, nxt);
    k2_chase<<<dim3(B), 256, 0, stream>>>(x, nxt, starts, nsegs,
                                          out + (size_t)B * T_FRAMES * DIMS);
    k3_means<<<dim3(T_FRAMES, B), 64, 0, stream>>>(x, starts, nsegs, out);
}